// Quantizer_75892072120910
// MI455X (gfx1250) — compile-verified
//
#include <hip/hip_runtime.h>
#include <hip/hip_bf16.h>
#include <math.h>

typedef float v2f __attribute__((ext_vector_type(2)));
typedef float v4f __attribute__((ext_vector_type(4)));
typedef float v8f __attribute__((ext_vector_type(8)));

#define DDIM 256
#define KCODES 1024
#define ROWS_PER_BLOCK 64
#define NTHREADS 256

// ---------------------------------------------------------------------------
// Kernel 0: normalize codebook W -> en (workspace), zero hist + loss sum.
// One block per code (K blocks, 256 threads = D).
// ---------------------------------------------------------------------------
__global__ __launch_bounds__(256) void vq_prep(const float* __restrict__ W,
                                               float* __restrict__ en,
                                               float* __restrict__ hist,
                                               float* __restrict__ losssum) {
    __shared__ float red[8];
    const int k = blockIdx.x;
    const int t = threadIdx.x;
    float w = W[k * DDIM + t];
    float s = w * w;
#pragma unroll
    for (int d = 16; d; d >>= 1) s += __shfl_down(s, d, 32);
    if ((t & 31) == 0) red[t >> 5] = s;
    __syncthreads();
    if (t == 0) {
        float tot = 0.f;
#pragma unroll
        for (int i = 0; i < 8; ++i) tot += red[i];
        red[0] = 1.0f / fmaxf(sqrtf(tot), 1e-12f);
        hist[k] = 0.f;
        if (k == 0) *losssum = 0.f;
    }
    __syncthreads();
    en[k * DDIM + t] = w * red[0];
}

// ---------------------------------------------------------------------------
// Kernel 1: main VQ kernel. One block = 64 rows. 8 waves; each wave streams
// 8 code-tiles (16 codes each) x 4 row-tiles using V_WMMA_F32_16X16X4_F32
// with running argmax, then writes quantized_st + one-hot encodings.
// ---------------------------------------------------------------------------
__global__ __launch_bounds__(256) void vq_main(const float* __restrict__ fin_g,
                                               const float* __restrict__ en,
                                               float* __restrict__ qout,
                                               float* __restrict__ enc,
                                               float* __restrict__ hist,
                                               float* __restrict__ losssum) {
    __shared__ __align__(16) float fn[ROWS_PER_BLOCK * DDIM];   // 64 KB normalized rows
    __shared__ float partial[NTHREADS];
    __shared__ float invn[ROWS_PER_BLOCK];
    __shared__ float wbest[8][ROWS_PER_BLOCK];
    __shared__ int   widx[8][ROWS_PER_BLOCK];
    __shared__ int   fidx[ROWS_PER_BLOCK];
    __shared__ float lloss;

    const int tid  = threadIdx.x;
    const int lane = tid & 31;
    const int wave = tid >> 5;
    const int l15  = lane & 15;
    const int half = lane >> 4;           // 0: K=0,1 / rows v ; 1: K=2,3 / rows v+8
    const int aoff = half * 2;
    const long long rbase = (long long)blockIdx.x * ROWS_PER_BLOCK;

    // ---- Stage 64 rows into LDS (coalesced float4) ----
    const v4f* fin4 = reinterpret_cast<const v4f*>(fin_g + rbase * DDIM);
    v4f* fn4 = reinterpret_cast<v4f*>(fn);
#pragma unroll
    for (int i = 0; i < 16; ++i) fn4[i * NTHREADS + tid] = fin4[i * NTHREADS + tid];
    __syncthreads();

    // ---- Row L2 norms: 4 threads per row ----
    {
        const int r = tid >> 2, q = tid & 3;
        const float* row = fn + r * DDIM + q * 64;
        float s = 0.f;
#pragma unroll
        for (int j = 0; j < 64; ++j) s += row[j] * row[j];
        partial[tid] = s;
    }
    __syncthreads();
    if (tid < ROWS_PER_BLOCK) {
        float s = partial[tid * 4] + partial[tid * 4 + 1] +
                  partial[tid * 4 + 2] + partial[tid * 4 + 3];
        invn[tid] = 1.0f / fmaxf(sqrtf(s), 1e-12f);
    }
    if (tid == 0) lloss = 0.f;
    __syncthreads();
#pragma unroll
    for (int i = 0; i < 16; ++i) {
        int fi = i * NTHREADS + tid;
        int r = fi >> 6;                 // 64 float4 per row
        v4f v = fn4[fi];
        v *= invn[r];
        fn4[fi] = v;
    }
    __syncthreads();

    // ---- WMMA scores + streaming argmax ----
    float best[4][8];
    int   bidx[4][8];
#pragma unroll
    for (int m = 0; m < 4; ++m)
#pragma unroll
        for (int v = 0; v < 8; ++v) { best[m][v] = -2.0f; bidx[m][v] = 0; }

    const v8f vzero = {0.f, 0.f, 0.f, 0.f, 0.f, 0.f, 0.f, 0.f};
    const float* aptr0 = fn + l15 * DDIM + aoff;

    for (int c = 0; c < 8; ++c) {
        const int c0 = (wave * 8 + c) * 16;
        const float* bptr = en + (long long)(c0 + l15) * DDIM + aoff;
        v8f acc[4];
#pragma unroll
        for (int m = 0; m < 4; ++m) acc[m] = vzero;
        for (int kk = 0; kk < 64; ++kk) {
            const int d0 = kk * 4;
            v2f b = *reinterpret_cast<const v2f*>(bptr + d0);
#pragma unroll
            for (int m = 0; m < 4; ++m) {
                v2f a = *reinterpret_cast<const v2f*>(aptr0 + m * 16 * DDIM + d0);
                acc[m] = __builtin_amdgcn_wmma_f32_16x16x4_f32(
                    false, a, false, b, (short)0, acc[m], false, false);
            }
        }
#pragma unroll
        for (int m = 0; m < 4; ++m)
#pragma unroll
            for (int v = 0; v < 8; ++v) {
                float s = acc[m][v];
                if (s > best[m][v]) { best[m][v] = s; bidx[m][v] = c0 + l15; }
            }
    }

    // ---- Cross-lane argmax within each 16-lane half ----
#pragma unroll
    for (int m = 0; m < 4; ++m)
#pragma unroll
        for (int v = 0; v < 8; ++v) {
            float b_ = best[m][v]; int i_ = bidx[m][v];
#pragma unroll
            for (int s = 1; s < 16; s <<= 1) {
                float ob = __shfl_xor(b_, s, 32);
                int   oi = __shfl_xor(i_, s, 32);
                if (ob > b_ || (ob == b_ && oi < i_)) { b_ = ob; i_ = oi; }
            }
            if (l15 == 0) {
                int row = m * 16 + v + half * 8;
                wbest[wave][row] = b_;
                widx[wave][row]  = i_;
            }
        }
    __syncthreads();

    // ---- Cross-wave argmax, histogram + loss ----
    if (tid < ROWS_PER_BLOCK) {
        float b_ = wbest[0][tid]; int i_ = widx[0][tid];
#pragma unroll
        for (int w = 1; w < 8; ++w) {
            float ob = wbest[w][tid]; int oi = widx[w][tid];
            if (ob > b_ || (ob == b_ && oi < i_)) { b_ = ob; i_ = oi; }
        }
        fidx[tid] = i_;
        atomicAdd(&hist[i_], 1.0f);
        atomicAdd(&lloss, 2.0f - 2.0f * b_);   // ||en[idx]-fn||^2 for this row
    }
    __syncthreads();
    if (tid == 0) atomicAdd(losssum, lloss);

    // ---- quantized_st = xn + (qn - xn) ----
    for (int r = 0; r < ROWS_PER_BLOCK; ++r) {
        float q = en[(long long)fidx[r] * DDIM + tid];
        float x = fn[r * DDIM + tid];
        __builtin_nontemporal_store(x + (q - x), qout + (rbase + r) * DDIM + tid);
    }

    // ---- one-hot encodings: one row (1024 floats) per iteration, float4 ----
    for (int r = 0; r < ROWS_PER_BLOCK; ++r) {
        v4f v = {0.f, 0.f, 0.f, 0.f};
        const int fi = fidx[r];
        if ((fi >> 2) == tid) v[fi & 3] = 1.0f;
        v4f* dst = reinterpret_cast<v4f*>(enc + (rbase + r) * KCODES) + tid;
        __builtin_nontemporal_store(v, dst);
    }
}

// ---------------------------------------------------------------------------
// Kernel 2: finalize scalars (perplexity + latent loss). 1 block, K threads.
// ---------------------------------------------------------------------------
__global__ __launch_bounds__(1024) void vq_finalize(const float* __restrict__ hist,
                                                    const float* __restrict__ losssum,
                                                    float* __restrict__ loss_out,
                                                    float* __restrict__ perp_out,
                                                    float invN, float invND) {
    __shared__ float red[32];
    const int t = threadIdx.x;
    float p = hist[t] * invN;
    float s = p * logf(p + 1e-10f);
#pragma unroll
    for (int d = 16; d; d >>= 1) s += __shfl_down(s, d, 32);
    if ((t & 31) == 0) red[t >> 5] = s;
    __syncthreads();
    if (t == 0) {
        float tot = 0.f;
#pragma unroll
        for (int i = 0; i < 32; ++i) tot += red[i];
        *perp_out = expf(-tot);
        *loss_out = 1.25f * (*losssum) * invND;
    }
}

// ---------------------------------------------------------------------------
extern "C" void kernel_launch(void* const* d_in, const int* in_sizes, int n_in,
                              void* d_out, int out_size, void* d_ws, size_t ws_size,
                              hipStream_t stream) {
    const float* f = (const float*)d_in[0];   // [16,8192,256]
    const float* W = (const float*)d_in[1];   // [1024,256]
    const int N = in_sizes[0] / DDIM;         // 131072
    const int K = in_sizes[1] / DDIM;         // 1024

    float* out      = (float*)d_out;
    float* qout     = out;                          // N*D
    float* loss_out = out + (size_t)N * DDIM;       // 1
    float* perp_out = loss_out + 1;                 // 1
    float* enc      = perp_out + 1;                 // N*K

    float* ws      = (float*)d_ws;
    float* en      = ws;                            // K*D normalized codebook
    float* hist    = en + (size_t)K * DDIM;         // K
    float* losssum = hist + K;                      // 1

    vq_prep<<<K, 256, 0, stream>>>(W, en, hist, losssum);
    vq_main<<<N / ROWS_PER_BLOCK, NTHREADS, 0, stream>>>(f, en, qout, enc, hist, losssum);
    vq_finalize<<<1, K, 0, stream>>>(hist, losssum, loss_out, perp_out,
                                     1.0f / (float)N, 1.0f / ((float)N * (float)DDIM));
}